// BiLSTM_CRF_79096117723905
// MI455X (gfx1250) — compile-verified
//
#include <hip/hip_runtime.h>
#include <math.h>

#define B_N   512
#define L_N   256
#define E_N   64
#define H_N   128
#define FH_N  512   // 4*H
#define K_N   18
#define KTOT  192   // E + H
#define NT_N  8     // H/16 hidden tiles
#define KK_N  48    // KTOT/4 k-steps
#define PACK_PER_DIR (NT_N * KK_N * 32 * 8)   // floats per direction

typedef float v2f __attribute__((ext_vector_type(2)));
typedef float v4f __attribute__((ext_vector_type(4)));
typedef float v8f __attribute__((ext_vector_type(8)));

__device__ __forceinline__ float sigmoidf_(float x) { return 1.0f / (1.0f + __expf(-x)); }

// One-shot repack of [Wx;Wh] (both directions) into per-(ntile, kstep, lane)
// WMMA B-fragment order: lane holds {i0,i1,f0,f1, g0,g1,o0,o1} contiguously,
// so the recurrence loop does two b128 loads instead of eight strided b32s.
__global__ __launch_bounds__(256) void pack_weights_kernel(
    const float* __restrict__ Wx_f, const float* __restrict__ Wh_f,
    const float* __restrict__ Wx_b, const float* __restrict__ Wh_b,
    float* __restrict__ pack)
{
  const int idx = blockIdx.x * blockDim.x + threadIdx.x;   // 2*8*48*32 = 24576
  if (idx >= 2 * NT_N * KK_N * 32) return;
  const int lane = idx & 31;
  const int kk   = (idx >> 5) % KK_N;
  const int nt   = ((idx >> 5) / KK_N) % NT_N;
  const int d    = idx / (NT_N * KK_N * 32);

  const int ln   = lane & 15;
  const int hi   = lane >> 4;
  const int colb = nt * 16 + ln;
  const int kb   = kk * 4 + hi * 2;            // K pair base for this lane

  const float* Wx = (d == 0) ? Wx_f : Wx_b;
  const float* Wh = (d == 0) ? Wh_f : Wh_b;
  const float* Wsel;
  int krow;
  if (kb < E_N) { Wsel = Wx; krow = kb; } else { Wsel = Wh; krow = kb - E_N; }
  const float* w0 = Wsel + (size_t)krow * FH_N + colb;
  const float* w1 = w0 + FH_N;

  float* o = pack + (size_t)idx * 8;
  v4f p0 = { w0[0 * H_N], w1[0 * H_N], w0[1 * H_N], w1[1 * H_N] };   // i, f
  v4f p1 = { w0[2 * H_N], w1[2 * H_N], w0[3 * H_N], w1[3 * H_N] };   // g, o
  *(v4f*)o       = p0;
  *(v4f*)(o + 4) = p1;
}

// One timestep of both LSTM directions, fully fused:
//   z = [x_t, h_prev] @ [Wx; Wh] + b   (fp32 WMMA 16x16x4, K = 192)
//   c = sig(f)*c + sig(i)*tanh(g);  h = sig(o)*tanh(c)
// grid: (B/16, H/16, 2 directions); block: 32 (one wave, EXEC all ones).
__global__ __launch_bounds__(32) void lstm_step_kernel(
    const int*   __restrict__ inputs,
    const float* __restrict__ Etab,
    const float* __restrict__ b_f, const float* __restrict__ b_b,
    const float* __restrict__ pack,
    float* __restrict__ h_f, float* __restrict__ h_b,
    float* __restrict__ c_f, float* __restrict__ c_b,
    int s)
{
  const int lane = threadIdx.x;          // 0..31
  const int m0 = blockIdx.x * 16;        // batch tile origin
  const int nt = blockIdx.y;             // hidden-unit tile
  const int n0 = nt * 16;
  const int d  = blockIdx.z;             // 0 = forward, 1 = backward

  const float* bv = (d == 0) ? b_f : b_b;
  float* hseq = (d == 0) ? h_f : h_b;
  float* cst  = (d == 0) ? c_f : c_b;

  const int t  = (d == 0) ? s : (L_N - 1 - s);   // time index this step writes
  const int tp = (d == 0) ? (t - 1) : (t + 1);   // previous-step time index

  const int ln = lane & 15;
  const int hi = lane >> 4;

  // A-matrix lane row (M = lane&15 for both half-waves)
  const int mA  = m0 + ln;
  const int tok = inputs[mA * L_N + t];

  // B/C/D lane column (N = lane&15)
  const int colb = n0 + ln;

  // Packed B-fragment stream for this (direction, n-tile): 8 floats/lane/k-step.
  const float* pk = pack + (size_t)(d * NT_N + nt) * (KK_N * 32 * 8) + lane * 8;

  // Seed accumulators with the bias (bias depends only on the column).
  v8f acc_i, acc_f, acc_g, acc_o;
  {
    const float bi = bv[0 * H_N + colb];
    const float bf = bv[1 * H_N + colb];
    const float bg = bv[2 * H_N + colb];
    const float bo = bv[3 * H_N + colb];
#pragma unroll
    for (int r = 0; r < 8; ++r) { acc_i[r] = bi; acc_f[r] = bf; acc_g[r] = bg; acc_o[r] = bo; }
  }

  // ---- Embedding region: K = 0..63, A row = E[token] (contiguous) ----
  const float* epA = Etab + (size_t)tok * E_N + hi * 2;
#pragma unroll
  for (int kk = 0; kk < E_N / 4; ++kk) {
    const v2f a  = *(const v2f*)(epA + kk * 4);
    const v4f p0 = *(const v4f*)(pk + (size_t)kk * 256);
    const v4f p1 = *(const v4f*)(pk + (size_t)kk * 256 + 4);
    const v2f bi_ = { p0[0], p0[1] };
    const v2f bf_ = { p0[2], p0[3] };
    const v2f bg_ = { p1[0], p1[1] };
    const v2f bo_ = { p1[2], p1[3] };
    acc_i = __builtin_amdgcn_wmma_f32_16x16x4_f32(false, a, false, bi_, (short)0, acc_i, false, false);
    acc_f = __builtin_amdgcn_wmma_f32_16x16x4_f32(false, a, false, bf_, (short)0, acc_f, false, false);
    acc_g = __builtin_amdgcn_wmma_f32_16x16x4_f32(false, a, false, bg_, (short)0, acc_g, false, false);
    acc_o = __builtin_amdgcn_wmma_f32_16x16x4_f32(false, a, false, bo_, (short)0, acc_o, false, false);
  }

  // ---- Hidden region: K = 64..191, A row = h_prev (zero at s == 0: skip) ----
  if (s > 0) {
    const float* hpA = hseq + ((size_t)tp * B_N + mA) * H_N + hi * 2;
    const float* pkh = pk + (size_t)(E_N / 4) * 256;
#pragma unroll
    for (int kk = 0; kk < H_N / 4; ++kk) {
      const v2f a  = *(const v2f*)(hpA + kk * 4);
      const v4f p0 = *(const v4f*)(pkh + (size_t)kk * 256);
      const v4f p1 = *(const v4f*)(pkh + (size_t)kk * 256 + 4);
      const v2f bi_ = { p0[0], p0[1] };
      const v2f bf_ = { p0[2], p0[3] };
      const v2f bg_ = { p1[0], p1[1] };
      const v2f bo_ = { p1[2], p1[3] };
      acc_i = __builtin_amdgcn_wmma_f32_16x16x4_f32(false, a, false, bi_, (short)0, acc_i, false, false);
      acc_f = __builtin_amdgcn_wmma_f32_16x16x4_f32(false, a, false, bf_, (short)0, acc_f, false, false);
      acc_g = __builtin_amdgcn_wmma_f32_16x16x4_f32(false, a, false, bg_, (short)0, acc_g, false, false);
      acc_o = __builtin_amdgcn_wmma_f32_16x16x4_f32(false, a, false, bo_, (short)0, acc_o, false, false);
    }
  }

  // ---- Gate nonlinearities + state update (C/D layout: M = r + 8*hi, N = lane&15) ----
  float* hout = hseq + (size_t)t * B_N * H_N;
#pragma unroll
  for (int r = 0; r < 8; ++r) {
    const int m = m0 + r + hi * 8;
    const int ci = m * H_N + colb;
    const float cold = (s == 0) ? 0.0f : cst[ci];
    const float zi = acc_i[r], zf = acc_f[r], zg = acc_g[r], zo = acc_o[r];
    const float cn = sigmoidf_(zf) * cold + sigmoidf_(zi) * tanhf(zg);
    const float hn = sigmoidf_(zo) * tanhf(cn);
    cst[ci]  = cn;
    hout[ci] = hn;
  }
}

// logits[l][b][k] = [h_fwd | h_bwd] @ Wd + bd   (K=18 -> plain VALU)
__global__ __launch_bounds__(256) void dense_kernel(
    const float* __restrict__ h_f, const float* __restrict__ h_b,
    const float* __restrict__ Wd, const float* __restrict__ bd,
    float* __restrict__ logits)
{
  const int idx = blockIdx.x * blockDim.x + threadIdx.x;
  if (idx >= L_N * B_N * K_N) return;
  const int k = idx % K_N;
  const int b = (idx / K_N) % B_N;
  const int l = idx / (K_N * B_N);
  const float* hf = h_f + ((size_t)l * B_N + b) * H_N;
  const float* hb = h_b + ((size_t)l * B_N + b) * H_N;
  float sum = bd[k];
  for (int j = 0; j < H_N; ++j) {
    sum += hf[j] * Wd[j * K_N + k];
    sum += hb[j] * Wd[(H_N + j) * K_N + k];
  }
  logits[idx] = sum;
}

// CRF log-likelihood, one thread per batch row.
__global__ __launch_bounds__(128) void crf_kernel(
    const int* __restrict__ labels, const float* __restrict__ logits,
    const float* __restrict__ T, float* __restrict__ out)
{
  const int b = blockIdx.x * blockDim.x + threadIdx.x;
  if (b >= B_N) return;
  const int* lab = labels + (size_t)b * L_N;

  int len = 0;
  for (int l = 0; l < L_N; ++l) len += (lab[l] != 0) ? 1 : 0;

  float unary = 0.0f, binary = 0.0f;
  for (int l = 0; l < L_N; ++l)
    if (l < len) unary += logits[(size_t)l * B_N * K_N + b * K_N + lab[l]];
  for (int l = 0; l + 1 < L_N; ++l)
    if (l + 1 < len) binary += T[lab[l] * K_N + lab[l + 1]];

  float alpha[K_N], na[K_N];
  for (int j = 0; j < K_N; ++j) alpha[j] = logits[b * K_N + j];  // t = 0

  for (int t = 1; t < L_N; ++t) {
    if (t < len) {
      const float* lg = logits + (size_t)t * B_N * K_N + b * K_N;
      for (int j = 0; j < K_N; ++j) {
        float mx = -INFINITY;
        for (int i = 0; i < K_N; ++i) mx = fmaxf(mx, alpha[i] + T[i * K_N + j]);
        float ss = 0.0f;
        for (int i = 0; i < K_N; ++i) ss += __expf(alpha[i] + T[i * K_N + j] - mx);
        na[j] = mx + __logf(ss) + lg[j];
      }
      for (int j = 0; j < K_N; ++j) alpha[j] = na[j];
    }
  }

  float mx = -INFINITY;
  for (int j = 0; j < K_N; ++j) mx = fmaxf(mx, alpha[j]);
  float ss = 0.0f;
  for (int j = 0; j < K_N; ++j) ss += __expf(alpha[j] - mx);
  out[b] = unary + binary - (mx + __logf(ss));
}

__global__ void copyT_kernel(const float* __restrict__ T, float* __restrict__ out) {
  const int i = blockIdx.x * blockDim.x + threadIdx.x;
  if (i < K_N * K_N) out[B_N + i] = T[i];
}

extern "C" void kernel_launch(void* const* d_in, const int* in_sizes, int n_in,
                              void* d_out, int out_size, void* d_ws, size_t ws_size,
                              hipStream_t stream) {
  (void)in_sizes; (void)n_in; (void)out_size; (void)ws_size;

  const int*   inputs = (const int*)  d_in[0];
  const int*   labels = (const int*)  d_in[1];
  const float* Etab   = (const float*)d_in[2];
  const float* Wx_f   = (const float*)d_in[3];
  const float* Wh_f   = (const float*)d_in[4];
  const float* b_f    = (const float*)d_in[5];
  const float* Wx_b   = (const float*)d_in[6];
  const float* Wh_b   = (const float*)d_in[7];
  const float* b_b    = (const float*)d_in[8];
  const float* Wd     = (const float*)d_in[9];
  const float* bd     = (const float*)d_in[10];
  const float* T      = (const float*)d_in[11];
  float* out = (float*)d_out;

  // Workspace layout (floats): h_fwd, h_bwd [L][B][H]; c_fwd, c_bwd [B][H];
  // logits [L][B][K]; packed weight fragments [2][8][48][32][8]
  float* ws = (float*)d_ws;
  float* h_f    = ws;
  float* h_b    = h_f + (size_t)L_N * B_N * H_N;
  float* c_f    = h_b + (size_t)L_N * B_N * H_N;
  float* c_b    = c_f + (size_t)B_N * H_N;
  float* logits = c_b + (size_t)B_N * H_N;
  float* pack   = logits + (size_t)L_N * B_N * K_N;

  const int pack_threads = 2 * NT_N * KK_N * 32;
  pack_weights_kernel<<<(pack_threads + 255) / 256, 256, 0, stream>>>(
      Wx_f, Wh_f, Wx_b, Wh_b, pack);

  const dim3 lstm_grid(B_N / 16, H_N / 16, 2);
  for (int s = 0; s < L_N; ++s) {
    lstm_step_kernel<<<lstm_grid, 32, 0, stream>>>(
        inputs, Etab, b_f, b_b, pack, h_f, h_b, c_f, c_b, s);
  }

  const int dense_total = L_N * B_N * K_N;
  dense_kernel<<<(dense_total + 255) / 256, 256, 0, stream>>>(h_f, h_b, Wd, bd, logits);

  crf_kernel<<<(B_N + 127) / 128, 128, 0, stream>>>(labels, logits, T, out);

  copyT_kernel<<<(K_N * K_N + 255) / 256, 256, 0, stream>>>(T, out);
}